// LocalGrouper_49331994362263
// MI455X (gfx1250) — compile-verified
//
#include <hip/hip_runtime.h>
#include <cstdint>
#include <cstddef>

// Problem constants (fixed by reference setup_inputs / module constants)
#define BB   4
#define NN   8192
#define CC   64
#define SS   2048     // NUM_POINT
#define KK   32       // NUM_SAMPLE
#define R2   0.04f    // RADIUS^2

typedef __attribute__((ext_vector_type(2))) float v2f;
typedef __attribute__((ext_vector_type(8))) float v8f;

// ---------------------------------------------------------------------------
// Kernel 1: Farthest Point Sampling. One block (1024 threads) per batch.
// Each lane owns 8 points' running min-distance in registers; per-iteration
// argmax via wave32 shfl_xor reduction + 32-entry LDS combine.
// Writes new_xyz (float) and fps_idx (as float values) into d_out.
// ---------------------------------------------------------------------------
#define FPS_TPB 1024
#define FPS_PPT (NN / FPS_TPB)   // 8

__global__ __launch_bounds__(FPS_TPB)
void fps_kernel(const float* __restrict__ xyz,
                float* __restrict__ new_xyz,
                float* __restrict__ fps_idx_f) {
    const int b   = blockIdx.x;
    const int tid = threadIdx.x;
    const float* __restrict__ x = xyz + (size_t)b * NN * 3;

    __shared__ float s_d[32];
    __shared__ int   s_n[32];
    __shared__ float s_last[3];

    float md[FPS_PPT];
#pragma unroll
    for (int j = 0; j < FPS_PPT; ++j) md[j] = 1e10f;

    if (tid == 0) {
        float x0 = x[0], y0 = x[1], z0 = x[2];
        s_last[0] = x0; s_last[1] = y0; s_last[2] = z0;
        fps_idx_f[(size_t)b * SS + 0] = 0.0f;
        new_xyz[((size_t)b * SS + 0) * 3 + 0] = x0;
        new_xyz[((size_t)b * SS + 0) * 3 + 1] = y0;
        new_xyz[((size_t)b * SS + 0) * 3 + 2] = z0;
    }
    __syncthreads();

    for (int i = 1; i < SS; ++i) {
        const float lx = s_last[0], ly = s_last[1], lz = s_last[2];
        float bd = -1.0f;
        int   bn = 0x7FFFFFFF;
#pragma unroll
        for (int j = 0; j < FPS_PPT; ++j) {
            const int n = tid + j * FPS_TPB;
            const float dx = x[n * 3 + 0] - lx;
            const float dy = x[n * 3 + 1] - ly;
            const float dz = x[n * 3 + 2] - lz;
            float d = dx * dx + dy * dy + dz * dz;
            md[j] = fminf(md[j], d);
            // argmax with first-occurrence (lowest index) tie-break
            if (md[j] > bd || (md[j] == bd && n < bn)) { bd = md[j]; bn = n; }
        }
        // wave32 reduction
#pragma unroll
        for (int off = 16; off > 0; off >>= 1) {
            float od = __shfl_xor(bd, off, 32);
            int   on = __shfl_xor(bn, off, 32);
            if (od > bd || (od == bd && on < bn)) { bd = od; bn = on; }
        }
        const int wave = tid >> 5;
        if ((tid & 31) == 0) { s_d[wave] = bd; s_n[wave] = bn; }
        __syncthreads();
        if (tid < 32) {
            bd = s_d[tid]; bn = s_n[tid];
#pragma unroll
            for (int off = 16; off > 0; off >>= 1) {
                float od = __shfl_xor(bd, off, 32);
                int   on = __shfl_xor(bn, off, 32);
                if (od > bd || (od == bd && on < bn)) { bd = od; bn = on; }
            }
            if (tid == 0) {
                fps_idx_f[(size_t)b * SS + i] = (float)bn;
                const float nx = x[bn * 3 + 0];
                const float ny = x[bn * 3 + 1];
                const float nz = x[bn * 3 + 2];
                new_xyz[((size_t)b * SS + i) * 3 + 0] = nx;
                new_xyz[((size_t)b * SS + i) * 3 + 1] = ny;
                new_xyz[((size_t)b * SS + i) * 3 + 2] = nz;
                s_last[0] = nx; s_last[1] = ny; s_last[2] = nz;
            }
        }
        __syncthreads();
    }
}

// ---------------------------------------------------------------------------
// Kernel 2: Ball query via V_WMMA_F32_16X16X4_F32.
// One wave per 16 centers. A = 16x4 (cx,cy,cz,1); per 16-point tile
// B = 4x16 (-2px,-2py,-2pz,|p|^2); D = -2 c.p + |p|^2, compared against
// thr = r^2 - |c|^2 per row. Ordered first-K compaction from ballot masks:
// C/D layout -> VGPR v holds row v (lanes 0-15) and row v+8 (lanes 16-31).
// ---------------------------------------------------------------------------
__global__ __launch_bounds__(128)
void ballq_kernel(const float* __restrict__ xyz,
                  const float* __restrict__ new_xyz,
                  int* __restrict__ idx_out) {
    const int wave = threadIdx.x >> 5;
    const int lane = threadIdx.x & 31;
    const int tile = blockIdx.x * (blockDim.x >> 5) + wave; // over B*S/16
    const int tilesPerBatch = SS / 16;
    const int b  = tile / tilesPerBatch;
    const int s0 = (tile % tilesPerBatch) * 16;

    const float* __restrict__ x = xyz + (size_t)b * NN * 3;
    const float* __restrict__ c = new_xyz + ((size_t)b * SS + s0) * 3;

    const int cl = lane & 15;
    const float cx = c[cl * 3 + 0];
    const float cy = c[cl * 3 + 1];
    const float cz = c[cl * 3 + 2];
    const float nc = cx * cx + cy * cy + cz * cz;

    // A-matrix 16x4 layout: VGPR0 = K0 (lanes0-15) / K2 (lanes16-31),
    //                       VGPR1 = K1 (lanes0-15) / K3 (lanes16-31)
    v2f A;
    A.x = (lane < 16) ? cx : cz;
    A.y = (lane < 16) ? cy : 1.0f;

    // Per-output-VGPR threshold: row = v + 8*(lane>=16)
    float thr[8];
#pragma unroll
    for (int v = 0; v < 8; ++v) {
        const int src = v + ((lane >> 4) << 3);
        thr[v] = R2 - __shfl(nc, src, 32);
    }

    const int row  = lane & 15;
    const int vsel = row & 7;
    const int hsh  = (row >> 3) * 16;
    int* __restrict__ myout = idx_out + (((size_t)b * SS + s0 + row) * KK);

    int cnt      = (lane < 16) ? 0 : KK;   // upper lanes report "done"
    int firstIdx = -1;

    for (int n0 = 0; n0 < NN; n0 += 16) {
        const float px = x[(n0 + cl) * 3 + 0];
        const float py = x[(n0 + cl) * 3 + 1];
        const float pz = x[(n0 + cl) * 3 + 2];
        // B-matrix 4x16 layout mirrors A: VGPR0 = K0/K2, VGPR1 = K1/K3
        v2f Bm;
        Bm.x = (lane < 16) ? (-2.0f * px) : (-2.0f * pz);
        Bm.y = (lane < 16) ? (-2.0f * py) : (px * px + py * py + pz * pz);

        v8f Cc = {};
        v8f D = __builtin_amdgcn_wmma_f32_16x16x4_f32(
            /*neg_a=*/false, A, /*neg_b=*/false, Bm,
            /*c_mod=*/(short)0, Cc, /*reuse_a=*/false, /*reuse_b=*/false);

        // Select this row's 16-bit in-ball mask without spilling
        unsigned mym = 0u;
#pragma unroll
        for (int v = 0; v < 8; ++v) {
            const unsigned mv = __builtin_amdgcn_ballot_w32(D[v] < thr[v]);
            if (vsel == v) mym = mv;
        }
        mym = (mym >> hsh) & 0xFFFFu;

        if (lane < 16 && cnt < KK) {
            while (mym && cnt < KK) {
                const int cbit = __builtin_ctz(mym);
                const int pid  = n0 + cbit;
                if (cnt == 0) firstIdx = pid;
                myout[cnt++] = pid;
                mym &= (mym - 1u);
            }
        }
        if (__builtin_amdgcn_ballot_w32(cnt < KK) == 0u) break;
    }

    if (lane < 16) {
        const int pad = (firstIdx < 0) ? 0 : firstIdx;
        for (int k = cnt; k < KK; ++k) myout[k] = pad;
    }
}

// ---------------------------------------------------------------------------
// Kernel 3: Gather + group. One wave per (b,s); lane = k. Writes
// grouped (B, 3+C, S, K): per channel a wave stores 32 contiguous floats
// (128B coalesced); feature gathers hit L2 (8MB footprint << 192MB L2).
// ---------------------------------------------------------------------------
__global__ __launch_bounds__(256)
void group_kernel(const float* __restrict__ xyz,
                  const float* __restrict__ feat,
                  const float* __restrict__ new_xyz,
                  const int* __restrict__ idx,
                  float* __restrict__ grouped) {
    const int wave = threadIdx.x >> 5;
    const int lane = threadIdx.x & 31;
    const int g = blockIdx.x * (blockDim.x >> 5) + wave; // over B*S
    const int b = g / SS;
    const int s = g % SS;

    const int pid = idx[((size_t)b * SS + s) * KK + lane];

    const float* __restrict__ p = xyz + ((size_t)b * NN + pid) * 3;
    const float px = p[0], py = p[1], pz = p[2];
    const float* __restrict__ c = new_xyz + ((size_t)b * SS + s) * 3;
    const float cx = c[0], cy = c[1], cz = c[2];

    const size_t chStride = (size_t)SS * KK;
    size_t base = (((size_t)b * (CC + 3)) * SS + s) * KK + lane;

    grouped[base]                = px - cx;
    grouped[base + chStride]     = py - cy;
    grouped[base + 2 * chStride] = pz - cz;

    const float* __restrict__ fb = feat + (size_t)b * CC * NN;
    base += 3 * chStride;
#pragma unroll 4
    for (int ch = 0; ch < CC; ++ch) {
        grouped[base + (size_t)ch * chStride] = fb[(size_t)ch * NN + pid];
    }
}

// ---------------------------------------------------------------------------
// Launch: d_in[0]=xyz (B,N,3) f32, d_in[1]=features (B,C,N) f32.
// d_out (float): [ new_xyz B*S*3 | grouped B*(3+C)*S*K | fps_idx B*S ].
// d_ws: ball-query index lists, B*S*K int32 (1 MB).
// ---------------------------------------------------------------------------
extern "C" void kernel_launch(void* const* d_in, const int* in_sizes, int n_in,
                              void* d_out, int out_size, void* d_ws, size_t ws_size,
                              hipStream_t stream) {
    const float* xyz  = (const float*)d_in[0];
    const float* feat = (const float*)d_in[1];

    float* out      = (float*)d_out;
    float* new_xyz  = out;                                         // B*S*3
    float* grouped  = out + (size_t)BB * SS * 3;                   // B*(3+C)*S*K
    float* fps_idx  = grouped + (size_t)BB * (CC + 3) * SS * KK;   // B*S (as float)

    int* idx_ws = (int*)d_ws;                                      // B*S*K ints

    // 1) FPS: one block per batch
    fps_kernel<<<BB, FPS_TPB, 0, stream>>>(xyz, new_xyz, fps_idx);

    // 2) Ball query: B*S/16 waves, 4 waves per block
    {
        const int tiles = BB * (SS / 16);       // 512
        const int waves_per_block = 4;
        ballq_kernel<<<tiles / waves_per_block, 32 * waves_per_block, 0, stream>>>(
            xyz, new_xyz, idx_ws);
    }

    // 3) Gather/group: one wave per (b,s), 8 waves per block
    {
        const int groups = BB * SS;             // 8192
        const int waves_per_block = 8;
        group_kernel<<<groups / waves_per_block, 32 * waves_per_block, 0, stream>>>(
            xyz, feat, new_xyz, idx_ws, grouped);
    }
}